// TabERA_19748259627486
// MI455X (gfx1250) — compile-verified
//
#include <hip/hip_runtime.h>

typedef __attribute__((ext_vector_type(2))) float v2f;
typedef __attribute__((ext_vector_type(8))) float v8f;

#define D_ 128
#define MAXG_ 2048
#define KSEL 16
#define CHUNK 512
#define NCHUNK (MAXG_ / CHUNK)        // 4
#define TILES_PER_CHUNK (CHUNK / 16)  // 32
#define NWAVES 8

// Order-preserving map f32 -> u32 (monotonic: bigger float => bigger key)
__device__ __forceinline__ unsigned int fkey(float f) {
  unsigned int b = __float_as_uint(f);
  return (b & 0x80000000u) ? ~b : (b | 0x80000000u);
}

// ---------- one wave per 128-float row: L2 norm ----------
__global__ __launch_bounds__(256)
void rownorm_kernel(const float* __restrict__ x, float* __restrict__ invn,
                    float* __restrict__ xn, int nrows) {
  int gwave = (blockIdx.x * blockDim.x + threadIdx.x) >> 5;
  int lane  = threadIdx.x & 31;
  if (gwave >= nrows) return;
  const float* row = x + (size_t)gwave * D_;
  float4 v = *(const float4*)(row + lane * 4);
  float s = v.x * v.x + v.y * v.y + v.z * v.z + v.w * v.w;
  #pragma unroll
  for (int o = 16; o > 0; o >>= 1) s += __shfl_xor(s, o, 32);
  float inv = 1.0f / fmaxf(sqrtf(s), 1e-12f);
  if (invn && lane == 0) invn[gwave] = inv;
  if (xn) {
    float4 r; r.x = v.x * inv; r.y = v.y * inv; r.z = v.z * inv; r.w = v.w * inv;
    *(float4*)(xn + (size_t)gwave * D_ + lane * 4) = r;
  }
}

// ---------- counting-sort queries by partition ----------
__global__ void zero_counts_kernel(int* counts, int P) {
  if ((int)threadIdx.x < P) counts[threadIdx.x] = 0;
}
__global__ void rank_kernel(const int* __restrict__ ha, int* counts,
                            int* ranks, int Bn) {
  int b = blockIdx.x * blockDim.x + threadIdx.x;
  if (b < Bn) ranks[b] = atomicAdd(&counts[ha[b]], 1);
}
__global__ void scan_kernel(const int* __restrict__ counts, int* offsets, int P) {
  if (threadIdx.x == 0) {
    int off = 0;
    for (int p = 0; p < P; ++p) { offsets[p] = off; off += counts[p]; }
  }
}
__global__ void scatter_kernel(const int* __restrict__ ha, const int* __restrict__ ranks,
                               const int* __restrict__ offsets, int* slots, int Bn) {
  int b = blockIdx.x * blockDim.x + threadIdx.x;
  if (b < Bn) slots[offsets[ha[b]] + ranks[b]] = b;
}

// ---------- main: sim via f32 WMMA + chunked top-16 ----------
__global__ __launch_bounds__(256)
void sim_topk_kernel(const float* __restrict__ qnorm,
                     const float* __restrict__ keys,
                     const float* __restrict__ vals,
                     const float* __restrict__ labels,
                     const int* __restrict__ cached_groups,
                     const int* __restrict__ group_sizes,
                     const int* __restrict__ counts,
                     const int* __restrict__ offsets,
                     const int* __restrict__ slots,
                     const float* __restrict__ invk,
                     const int* __restrict__ kptr,
                     float* __restrict__ out,
                     int Bn, int Nn) {
  __shared__ int candLds[MAXG_];                       // 8 KB
  __shared__ float simLds[16 * CHUNK];                 // 32 KB
  __shared__ unsigned long long run[2][16][KSEL];      // 4 KB

  const int p  = blockIdx.x;
  const int qt = blockIdx.y;
  const int Qp = counts[p];
  if (qt * 16 >= Qp) return;
  const int off = offsets[p];

  const int tid  = threadIdx.x;
  const int lane = tid & 31;
  const int wave = tid >> 5;

  // stage candidate index list (shared by all 16 queries of this partition)
  const int* cg = cached_groups + (size_t)p * MAXG_;
  for (int i = tid; i < MAXG_; i += 256) candLds[i] = cg[i];
  run[0][tid >> 4][tid & 15] = 0ull;                   // 256 entries, 256 threads

  // A fragments: 16x4 f32 tile layout -> lane holds row M=lane%16,
  // K pair = {0,1} (lanes 0-15) or {2,3} (lanes 16-31). 32 K-steps cover D=128.
  const int mrow = lane & 15;
  const int koff = (lane < 16) ? 0 : 2;
  int qs0   = qt * 16 + mrow;
  int qid_m = slots[off + (qs0 < Qp ? qs0 : 0)];
  v2f aF[32];
  const float* qrow = qnorm + (size_t)qid_m * D_;
  #pragma unroll
  for (int kk = 0; kk < 32; ++kk) aF[kk] = *(const v2f*)(qrow + kk * 4 + koff);

  __syncthreads();

  for (int chunk = 0; chunk < NCHUNK; ++chunk) {
    // ---- compute 16x16 sim tiles for this chunk of 512 candidates ----
    for (int i = 0; i < TILES_PER_CHUNK / NWAVES; ++i) {
      int tl = wave + i * NWAVES;                      // local tile 0..31
      int t  = chunk * TILES_PER_CHUNK + tl;
      int raw = candLds[t * 16 + mrow];
      int idx = raw < 0 ? 0 : raw;
      float inn = raw < 0 ? 0.0f : invk[idx];          // B=0 for invalid lanes
      const float* krow = keys + (size_t)idx * D_;
      __builtin_prefetch(krow, 0, 3);                  // global_prefetch_b8
      v8f c8 = {};
      #pragma unroll
      for (int kk = 0; kk < 32; ++kk) {
        // B tile 4x16: lane holds col N=lane%16, K pair {0,1}/{2,3} -> same koff
        v2f b2 = *(const v2f*)(krow + kk * 4 + koff);
        b2 *= inn;
        c8 = __builtin_amdgcn_wmma_f32_16x16x4_f32(
            /*neg_a=*/false, aF[kk], /*neg_b=*/false, b2,
            /*c_mod=*/(short)0, c8, /*reuse_a=*/false, /*reuse_b=*/false);
      }
      // C/D layout: VGPR j -> M = j + (lane<16?0:8), N = lane%16
      int mbase = (lane < 16) ? 0 : 8;
      bool valid = raw >= 0;
      #pragma unroll
      for (int j = 0; j < 8; ++j) {
        float v = valid ? c8[j] : -1000000000.0f;
        simLds[(mbase + j) * CHUNK + tl * 16 + mrow] = v;
      }
    }
    __syncthreads();

    // ---- merge-select top-16 per query: running top-16 + 512 fresh ----
    for (int m = wave; m < 16; m += NWAVES) {
      unsigned long long mine = (lane < KSEL) ? run[chunk & 1][m][lane] : 0ull;
      const float* srow = simLds + m * CHUNK;
      for (int j = 0; j < KSEL; ++j) {
        unsigned long long best = mine;
        for (int cc = lane; cc < CHUNK; cc += 32) {
          float v = srow[cc];
          unsigned int c = (unsigned int)(chunk * CHUNK + cc);
          unsigned long long key =
              ((unsigned long long)fkey(v) << 32) | (0xFFFFFFFFu - c);
          best = key > best ? key : best;
        }
        #pragma unroll
        for (int o = 16; o > 0; o >>= 1) {
          unsigned long long other = __shfl_xor(best, o, 32);
          best = other > best ? other : best;
        }
        if (mine == best) mine = 0ull;                 // winner was a carry-over
        unsigned int cbest = 0xFFFFFFFFu - (unsigned int)(best & 0xFFFFFFFFu);
        if ((best >> 32) != 0ull && cbest >= (unsigned int)(chunk * CHUNK)) {
          if (lane == 0)                               // winner from this chunk
            simLds[m * CHUNK + (cbest - chunk * CHUNK)] = -3.0e38f;
        }
        if (lane == 0) run[(chunk + 1) & 1][m][j] = best;
      }
    }
    __syncthreads();
  }

  // ---- emit outputs (final sorted top-16 sits in run[0]) ----
  const int kval = kptr[0];
  const bool normal = group_sizes[p] >= kval;
  float* outNK = out;
  float* outNV = out + (size_t)Bn * KSEL * D_;
  float* outNL = out + 2 * (size_t)Bn * KSEL * D_;
  float* outIX = outNL + (size_t)Bn * KSEL;

  for (int m = wave; m < 16; m += NWAVES) {
    int qs = qt * 16 + m;
    if (qs >= Qp) continue;
    int qid = slots[off + qs];
    for (int j = 0; j < KSEL; ++j) {
      unsigned long long key = run[NCHUNK & 1][m][j];
      unsigned int cbest = 0xFFFFFFFFu - (unsigned int)(key & 0xFFFFFFFFu);
      int raw  = ((key >> 32) != 0ull && cbest < MAXG_) ? candLds[cbest] : -1;
      int ifin = raw < 0 ? 0 : (raw < Nn ? raw : Nn - 1);
      size_t base = ((size_t)qid * KSEL + j) * D_;
      float4 kq, vq;
      if (normal) {
        kq = *(const float4*)(keys + (size_t)ifin * D_ + lane * 4);
        vq = *(const float4*)(vals + (size_t)ifin * D_ + lane * 4);
      } else {
        kq = make_float4(0.f, 0.f, 0.f, 0.f);
        vq = kq;
      }
      *(float4*)(outNK + base + lane * 4) = kq;
      *(float4*)(outNV + base + lane * 4) = vq;
      if (lane == 0) {
        outNL[(size_t)qid * KSEL + j] = normal ? labels[ifin] : 0.0f;
        outIX[(size_t)qid * KSEL + j] = normal ? (float)ifin : 0.0f;
      }
    }
  }
}

extern "C" void kernel_launch(void* const* d_in, const int* in_sizes, int n_in,
                              void* d_out, int out_size, void* d_ws, size_t ws_size,
                              hipStream_t stream) {
  const float* query  = (const float*)d_in[0];
  const float* keys   = (const float*)d_in[1];
  const float* vals   = (const float*)d_in[2];
  const float* labels = (const float*)d_in[3];
  const int*   ha     = (const int*)d_in[4];
  const int*   cg     = (const int*)d_in[5];
  const int*   gs     = (const int*)d_in[6];
  const int*   kptr   = (const int*)d_in[7];

  const int Bn = in_sizes[4];       // 1024
  const int Nn = in_sizes[3];       // 131072
  const int P  = in_sizes[6];       // 64

  // workspace layout
  unsigned char* ws = (unsigned char*)d_ws;
  int* counts  = (int*)(ws + 0);
  int* offsets = (int*)(ws + 256);
  int* ranks   = (int*)(ws + 512);
  int* slots   = (int*)(ws + 512 + (size_t)Bn * 4);
  size_t o = 512 + (size_t)Bn * 8;
  o = (o + 255) & ~(size_t)255;
  float* qnorm = (float*)(ws + o);
  o += (size_t)Bn * D_ * sizeof(float);
  float* invk  = (float*)(ws + o);

  zero_counts_kernel<<<1, 64, 0, stream>>>(counts, P);
  rownorm_kernel<<<(Nn + 7) / 8, 256, 0, stream>>>(keys, invk, nullptr, Nn);
  rownorm_kernel<<<(Bn + 7) / 8, 256, 0, stream>>>(query, nullptr, qnorm, Bn);
  rank_kernel<<<(Bn + 255) / 256, 256, 0, stream>>>(ha, counts, ranks, Bn);
  scan_kernel<<<1, 64, 0, stream>>>(counts, offsets, P);
  scatter_kernel<<<(Bn + 255) / 256, 256, 0, stream>>>(ha, ranks, offsets, slots, Bn);

  dim3 grid(P, (Bn + 15) / 16);
  sim_topk_kernel<<<grid, 256, 0, stream>>>(qnorm, keys, vals, labels, cg, gs,
                                            counts, offsets, slots, invk, kptr,
                                            (float*)d_out, Bn, Nn);
}